// PariwiseCorrelationDecoder_53996328845639
// MI455X (gfx1250) — compile-verified
//
#include <hip/hip_runtime.h>

typedef float v2f __attribute__((ext_vector_type(2)));
typedef float v4f __attribute__((ext_vector_type(4)));
typedef float v8f __attribute__((ext_vector_type(8)));

// score[b] = sum_{i,j,d} U[b,i,d] * V[b,j,d] * W[i,j],  K=D=8.
// Two batch elements (b0=2p, b1=2p+1) packed into one 16x16 WMMA output:
//   A(16x8) = [U_b0 ; U_b1]     (M = i + 8*bsel, K = d)
//   B(8x16) = [V_b0^T | V_b1^T] (K = d, N = j + 8*bsel)
// via two chained V_WMMA_F32_16X16X4_F32 (exact fp32).
// D layout (ISA 7.12.2): VGPR r, lanes 0-15 -> D[M=r][N=lane];
//                        lanes 16-31 -> D[M=r+8][N=lane-16].
// => H_b0 lives in lanes 0-7, H_b1 in lanes 24-31; cross blocks discarded.

// DPP8 xor-shuffles within each aligned group of 8 lanes (VALU-only).
__device__ __forceinline__ constexpr unsigned dpp8_sel(int a, int b, int c, int d,
                                                       int e, int f, int g, int h) {
    return (unsigned)(a | (b << 3) | (c << 6) | (d << 9) |
                      (e << 12) | (f << 15) | (g << 18) | (h << 21));
}
__device__ __forceinline__ float xor_add_dpp8_1(float s) {
    constexpr unsigned SEL = dpp8_sel(1, 0, 3, 2, 5, 4, 7, 6);  // i ^ 1
    return s + __int_as_float(__builtin_amdgcn_mov_dpp8(__float_as_int(s), SEL));
}
__device__ __forceinline__ float xor_add_dpp8_2(float s) {
    constexpr unsigned SEL = dpp8_sel(2, 3, 0, 1, 6, 7, 4, 5);  // i ^ 2
    return s + __int_as_float(__builtin_amdgcn_mov_dpp8(__float_as_int(s), SEL));
}
__device__ __forceinline__ float xor_add_dpp8_4(float s) {
    constexpr unsigned SEL = dpp8_sel(4, 5, 6, 7, 0, 1, 2, 3);  // i ^ 4
    return s + __int_as_float(__builtin_amdgcn_mov_dpp8(__float_as_int(s), SEL));
}

__global__ __launch_bounds__(256) void pairwise_corr_wmma_kernel(
    const float* __restrict__ u,   // [B,8,8]
    const float* __restrict__ v,   // [B,8,8]
    const float* __restrict__ w,   // [64] = W[8][8] row-major
    float* __restrict__ out,       // [B]
    int npairs)                    // B/2
{
    __shared__ float lds[8 * 256];          // 8 waves * (128 U + 128 V) floats
    const int lane = threadIdx.x & 31;
    // Force wave-uniform values into SGPRs: scalar loop counter, scalar
    // addressing (SADDR-form loads), scalar loop/prefetch control.
    const int waveInBlk = __builtin_amdgcn_readfirstlane((int)(threadIdx.x >> 5));
    float* ldsU = &lds[waveInBlk * 256];
    float* ldsV = ldsU + 128;

    // Per-lane operand gather indices (constant across iterations).
    const int half = lane >> 4;             // 0: K pair (0,1); 1: K pair (2,3)
    const int lsub = lane & 15;             // M for A, N for B
    const int bsel = lsub >> 3;             // which batch element of the pair
    const int rowj = lsub & 7;              // i (A) / j (B)
    const int d0   = 2 * half;              // step-0 starting d for this lane
    const int gidx = bsel * 64 + rowj * 8 + d0;   // same formula for A and B

    // Per-lane weight column: wl[r] = W[r][lane%8]  (L2-resident, 256 B total)
    float wl[8];
    #pragma unroll
    for (int r = 0; r < 8; ++r) wl[r] = w[r * 8 + (lane & 7)];

    const int nWavesBlk = (int)(blockDim.x >> 5);
    const int waveId =
        __builtin_amdgcn_readfirstlane((int)blockIdx.x * nWavesBlk + waveInBlk);
    const int nWaves = (int)gridDim.x * nWavesBlk;

    for (int p = waveId; p < npairs; p += nWaves) {
        const size_t base = (size_t)p * 128;

        // Stream 1 KB (two batch elements of U and V), read-once -> NT hint.
        v4f u4 = __builtin_nontemporal_load((const v4f*)(u + base) + lane);
        v4f v4_ = __builtin_nontemporal_load((const v4f*)(v + base) + lane);

        // Branch-free prefetch of next grid-stride iteration (scalar clamp:
        // last iteration re-prefetches its own lines, which is harmless).
        int pnext = p + nWaves;
        pnext = (pnext < npairs) ? pnext : p;
        const size_t nbase = (size_t)pnext * 128;
        __builtin_prefetch(u + nbase + lane * 4, 0, 3);
        __builtin_prefetch(v + nbase + lane * 4, 0, 3);

        // Stage into wave-private LDS (no barrier: same-wave DS ops in-order).
        *(v4f*)&ldsU[lane * 4] = u4;
        *(v4f*)&ldsV[lane * 4] = v4_;

        // Gather WMMA operands: merged b64 LDS loads per operand per K-step.
        v2f a0 = *(const v2f*)&ldsU[gidx];        // step 0: d in {0,1}/{2,3}
        v2f b0 = *(const v2f*)&ldsV[gidx];
        v2f a1 = *(const v2f*)&ldsU[gidx + 4];    // step 1: d in {4,5}/{6,7}
        v2f b1 = *(const v2f*)&ldsV[gidx + 4];

        // D = A x B over K=8 via two K=4 fp32 WMMAs (exact).
        v8f c = {};
        c = __builtin_amdgcn_wmma_f32_16x16x4_f32(
                false, a0, false, b0, (short)0, c, false, false);
        c = __builtin_amdgcn_wmma_f32_16x16x4_f32(
                false, a1, false, b1, (short)0, c, false, false);

        // Weighted contraction with W: lane holds column N of H, VGPR r = row.
        float s = c[0] * wl[0];
        s += c[1] * wl[1];
        s += c[2] * wl[2];
        s += c[3] * wl[3];
        s += c[4] * wl[4];
        s += c[5] * wl[5];
        s += c[6] * wl[6];
        s += c[7] * wl[7];

        // 8-lane xor-tree reduce, pure VALU via DPP8.
        // Afterwards: lanes 0-7 all hold score(b0), lanes 24-31 all score(b1).
        s = xor_add_dpp8_1(s);
        s = xor_add_dpp8_2(s);
        s = xor_add_dpp8_4(s);

        // Pull score(b1) to an SGPR (v_readlane ignores EXEC), then lane 0
        // issues one aligned 8-byte NT store of both scores.
        float sb1 = __int_as_float(
            __builtin_amdgcn_readlane(__float_as_int(s), 24));
        if (lane == 0) {
            v2f o;
            o.x = s;                       // score(b0)
            o.y = sb1;                     // score(b1)
            __builtin_nontemporal_store(o, (v2f*)(out + 2 * (size_t)p));
        }
    }
}

extern "C" void kernel_launch(void* const* d_in, const int* in_sizes, int n_in,
                              void* d_out, int out_size, void* d_ws, size_t ws_size,
                              hipStream_t stream) {
    const float* u = (const float*)d_in[0];   // [B,8,8] f32
    const float* v = (const float*)d_in[1];   // [B,8,8] f32
    const float* w = (const float*)d_in[2];   // [64]    f32
    float* out = (float*)d_out;               // [B]     f32

    const int npairs = in_sizes[0] / 128;     // B/2 (B = in_sizes[0]/64, even)

    const int threads   = 256;                // 8 wave32 waves per block
    const int wavesPerB = threads / 32;
    int blocks = 4096;                        // 32768 waves -> 32 pairs/wave
    if (blocks * wavesPerB > npairs) {
        blocks = (npairs + wavesPerB - 1) / wavesPerB;
        if (blocks < 1) blocks = 1;
    }

    pairwise_corr_wmma_kernel<<<blocks, threads, 0, stream>>>(u, v, w, out, npairs);
}